// SS2D_77506979824126
// MI455X (gfx1250) — compile-verified
//
#include <hip/hip_runtime.h>
#include <math.h>

// ---------------------------------------------------------------------------
// SS2D (VMamba) block for MI455X / gfx1250.
//  B=4, H=W=32, DM=192, DI=384, N=16, K=4 dirs, R=12, L=1024.
//  GEMMs use native fp32 WMMA (v_wmma_f32_16x16x4_f32, wave32, one wave per
//  16x16 tile).  Selective scan: 6144 independent (b,k,d) recurrences, state
//  in VGPRs; the shared (dt_r, B, C) 44x32 tile per chunk is staged into LDS
//  by the Tensor Data Mover (tensor_load_to_lds + s_wait_tensorcnt).
// ---------------------------------------------------------------------------

#define BB 4
#define HH 32
#define WW 32
#define DMO 192              // d_model
#define DI 384               // d_inner
#define NS 16                // d_state
#define KD 4                 // scan directions
#define RR 12                // dt_rank
#define LL (HH * WW)         // 1024
#define CPAD 48              // 44 (R+2N) padded to 3 WMMA tiles
#define LN_EPS 1e-5f
#define SCAN_CHUNK 32

typedef __attribute__((ext_vector_type(2))) float v2f;
typedef __attribute__((ext_vector_type(8))) float v8f;
typedef __attribute__((ext_vector_type(4))) unsigned int u32x4;
typedef __attribute__((ext_vector_type(8))) unsigned int u32x8;

__device__ __forceinline__ float siluf(float v) { return v / (1.0f + __expf(-v)); }
__device__ __forceinline__ float softplusf(float v) {
    return (v > 20.0f) ? v : log1pf(__expf(v));
}

// scan position j of direction k -> row-major spatial index
__device__ __forceinline__ int scan_map(int k, int j) {
    int jj = (k >= 2) ? (LL - 1 - j) : j;
    if (k & 1) jj = (jj & (HH - 1)) * WW + (jj >> 5);   // col-major transpose
    return jj;
}

// ---------------------------------------------------------------------------
// Kernel 1: in_proj GEMM  (4096 x 192) @ (192 x 768) -> split xi / silu(z)
// ---------------------------------------------------------------------------
__global__ __launch_bounds__(256) void k_inproj(const float* __restrict__ X,
                                                const float* __restrict__ Wp,
                                                float* __restrict__ xi,
                                                float* __restrict__ zs) {
    const int lane = threadIdx.x & 31;
    const int gw = blockIdx.x * 8 + (threadIdx.x >> 5);
    const int tm = gw / 48;                 // 768/16 = 48 N-tiles
    const int tn = gw % 48;
    const int mrow = tm * 16 + (lane & 15);
    const int ncol = tn * 16 + (lane & 15);
    const int khalf = (lane >> 4) * 2;
    const float* xrow = X + mrow * DMO;
    const float* wrow = Wp + ncol * DMO;    // B[k][n] = W[n][k]
    v8f acc = {};
    for (int kk = 0; kk < DMO; kk += 4) {
        v2f a, b;
        a.x = xrow[kk + khalf];
        a.y = xrow[kk + khalf + 1];
        b.x = wrow[kk + khalf];
        b.y = wrow[kk + khalf + 1];
        acc = __builtin_amdgcn_wmma_f32_16x16x4_f32(false, a, false, b,
                                                    (short)0, acc, false, false);
    }
    const int mbase = tm * 16 + ((lane >> 4) << 3);
#pragma unroll
    for (int r = 0; r < 8; ++r) {
        const int m = mbase + r;
        const float v = acc[r];
        if (ncol < DI) xi[m * DI + ncol] = v;
        else           zs[m * DI + (ncol - DI)] = siluf(v);
    }
}

// ---------------------------------------------------------------------------
// Kernel 2: depthwise 3x3 conv + bias + silu.  xi (B,L,DI) -> xc (B,DI,L)
// ---------------------------------------------------------------------------
__global__ void k_conv(const float* __restrict__ xi, const float* __restrict__ cw,
                       const float* __restrict__ cb, float* __restrict__ xc) {
    const int idx = blockIdx.x * blockDim.x + threadIdx.x;
    if (idx >= BB * DI * LL) return;
    const int s = idx % LL;
    const int d = (idx / LL) % DI;
    const int b = idx / (DI * LL);
    const int h = s >> 5, w = s & 31;
    float acc = cb[d];
#pragma unroll
    for (int kh = -1; kh <= 1; ++kh) {
        const int h2 = h + kh;
        if ((unsigned)h2 >= HH) continue;
#pragma unroll
        for (int kw = -1; kw <= 1; ++kw) {
            const int w2 = w + kw;
            if ((unsigned)w2 >= WW) continue;
            acc += cw[d * 9 + (kh + 1) * 3 + (kw + 1)] *
                   xi[(b * LL + h2 * WW + w2) * DI + d];
        }
    }
    xc[idx] = siluf(acc);
}

// ---------------------------------------------------------------------------
// Kernel 3: x_dbl GEMM per (b,k): (44x384) @ xs(384 x L); CrossScan index
// permutation folded into the B-fragment gather.  M padded to 48.
// ---------------------------------------------------------------------------
__global__ __launch_bounds__(256) void k_xdbl(const float* __restrict__ xpw,
                                              const float* __restrict__ xc,
                                              float* __restrict__ xdbl) {
    const int lane = threadIdx.x & 31;
    const int gw = blockIdx.x * 8 + (threadIdx.x >> 5);
    const int bk = gw / (3 * 64);
    const int t = gw % (3 * 64);
    const int tm = t / 64, tn = t % 64;
    const int b = bk >> 2, k = bk & 3;
    const int m = tm * 16 + (lane & 15);
    const int l = tn * 16 + (lane & 15);
    const int sidx = scan_map(k, l);
    const int khalf = (lane >> 4) * 2;
    const float* arow = xpw + (k * 44 + m) * DI;
    const float* bcol = xc + b * DI * LL + sidx;
    v8f acc = {};
    for (int kk = 0; kk < DI; kk += 4) {
        v2f a, bf;
        if (m < 44) { a.x = arow[kk + khalf]; a.y = arow[kk + khalf + 1]; }
        else        { a.x = 0.0f; a.y = 0.0f; }
        bf.x = bcol[(kk + khalf) * LL];
        bf.y = bcol[(kk + khalf + 1) * LL];
        acc = __builtin_amdgcn_wmma_f32_16x16x4_f32(false, a, false, bf,
                                                    (short)0, acc, false, false);
    }
    const int mbase = tm * 16 + ((lane >> 4) << 3);
#pragma unroll
    for (int r = 0; r < 8; ++r)
        xdbl[(bk * CPAD + (mbase + r)) * LL + l] = acc[r];
}

// ---------------------------------------------------------------------------
// Kernel 4: selective scan.  48 blocks (= B*K*3 d-segments) x 128 threads.
// Each chunk, wave 0 programs the Tensor Data Mover with a 2D D# that pulls
// the strided 44x32 (dt_r|B|C) tile into LDS; all waves consume after a
// tensorcnt wait + barrier.  dt GEMM (K=12) fused as 12 FMAs per step.
// ---------------------------------------------------------------------------
__global__ __launch_bounds__(128) void k_scan(const float* __restrict__ xdbl,
                                              const float* __restrict__ xc,
                                              const float* __restrict__ dtw,
                                              const float* __restrict__ dtb,
                                              const float* __restrict__ alog,
                                              const float* __restrict__ Dsp,
                                              float* __restrict__ ys) {
    __shared__ float lx[44 * SCAN_CHUNK];
    const int bk = blockIdx.x / 3;
    const int dseg = blockIdx.x % 3;
    const int b = bk >> 2, k = bk & 3;
    const int d = dseg * 128 + threadIdx.x;
    const int row = k * DI + d;

    float dw[RR];
#pragma unroll
    for (int r = 0; r < RR; ++r) dw[r] = dtw[row * RR + r];
    const float db = dtb[row];
    float Areg[NS];
#pragma unroll
    for (int n = 0; n < NS; ++n) Areg[n] = -__expf(alog[row * NS + n]);
    const float Dp = Dsp[row];
    float hst[NS];
#pragma unroll
    for (int n = 0; n < NS; ++n) hst[n] = 0.0f;

    const float* xcd = xc + (b * DI + d) * LL;
    const float* xd = xdbl + bk * CPAD * LL;
    float* ysd = ys + (bk * DI + d) * LL;
    const unsigned int lds_base = (unsigned int)(uintptr_t)&lx[0];

    for (int l0 = 0; l0 < LL; l0 += SCAN_CHUNK) {
        __syncthreads();                    // previous chunk fully consumed
        if (threadIdx.x < 32) {             // wave 0 programs the TDM
            const unsigned long long ga = (unsigned long long)(uintptr_t)(xd + l0);
            u32x4 g0;
            g0[0] = 1u;                                     // count=1, user D#
            g0[1] = lds_base;                               // lds_addr
            g0[2] = (unsigned int)ga;                       // global_addr[31:0]
            g0[3] = (unsigned int)((ga >> 32) & 0x1FFFFFFu) // global_addr[56:32]
                    | 0x80000000u;                          // type=2 (image)
            u32x8 g1;
            g1[0] = 2u << 16;                    // data_size = 4B; no multicast
            g1[1] = ((unsigned int)LL) << 16;    // tensor_dim0 = 1024 (lo16)
            g1[2] = 44u << 16;                   // tensor_dim0 hi=0; tensor_dim1=44
            g1[3] = ((unsigned int)SCAN_CHUNK) << 16; // tensor_dim1 hi=0; tile_dim0=32
            g1[4] = 44u;                         // tile_dim1 = 44; tile_dim2 = 0
            g1[5] = (unsigned int)LL;            // tensor_dim0_stride = 1024 (lo32)
            g1[6] = 0u;                          // stride hi16; dim1_stride lo16
            g1[7] = 0u;                          // dim1_stride hi32
            asm volatile("tensor_load_to_lds %0, %1"
                         :
                         : "s"(g0), "s"(g1)
                         : "memory");
            __builtin_amdgcn_s_wait_tensorcnt(0);
        }
        __syncthreads();                    // publish LDS tile to all waves
        for (int li = 0; li < SCAN_CHUNK; ++li) {
            const int l = l0 + li;
            const float u = xcd[scan_map(k, l)];
            float dr = db;
#pragma unroll
            for (int r = 0; r < RR; ++r) dr += dw[r] * lx[r * SCAN_CHUNK + li];
            const float delta = softplusf(dr);
            const float du = delta * u;
            float yv = 0.0f;
#pragma unroll
            for (int n = 0; n < NS; ++n) {
                const float Bv = lx[(RR + n) * SCAN_CHUNK + li];
                const float Cv = lx[(RR + NS + n) * SCAN_CHUNK + li];
                hst[n] = __expf(delta * Areg[n]) * hst[n] + du * Bv;
                yv += hst[n] * Cv;
            }
            ysd[l] = yv + Dp * u;
        }
    }
}

// ---------------------------------------------------------------------------
// Kernel 5: CrossMerge + LayerNorm + gate.  One block per (b, spatial l).
// ---------------------------------------------------------------------------
__global__ __launch_bounds__(128) void k_merge_ln(const float* __restrict__ ys,
                                                  const float* __restrict__ zs,
                                                  const float* __restrict__ lng,
                                                  const float* __restrict__ lnb,
                                                  float* __restrict__ yg) {
    __shared__ float red[128];
    __shared__ float red2[128];
    __shared__ float mu_s, rstd_s;
    const int b = blockIdx.x / LL;
    const int s = blockIdx.x % LL;
    const int h = s >> 5, w = s & 31;
    const int j1 = w * HH + h;            // scan pos of (h,w) in col-major dir

    float v[3];
    float sum = 0.0f, sq = 0.0f;
#pragma unroll
    for (int i = 0; i < 3; ++i) {
        const int d = threadIdx.x + i * 128;
        const float* base = ys + ((b * 4) * DI + d) * LL;
        const float y = base[s]                                   // k=0 fwd
                      + base[(2 * DI) * LL + (LL - 1 - s)]        // k=2 rev
                      + base[(1 * DI) * LL + j1]                  // k=1 colmaj
                      + base[(3 * DI) * LL + (LL - 1 - j1)];      // k=3 rev col
        v[i] = y;
        sum += y;
        sq += y * y;
    }
    red[threadIdx.x] = sum;
    red2[threadIdx.x] = sq;
    __syncthreads();
    for (int off = 64; off > 0; off >>= 1) {
        if (threadIdx.x < off) {
            red[threadIdx.x] += red[threadIdx.x + off];
            red2[threadIdx.x] += red2[threadIdx.x + off];
        }
        __syncthreads();
    }
    if (threadIdx.x == 0) {
        const float mu = red[0] / (float)DI;
        mu_s = mu;
        rstd_s = rsqrtf(red2[0] / (float)DI - mu * mu + LN_EPS);
    }
    __syncthreads();
    const float mu = mu_s, rstd = rstd_s;
#pragma unroll
    for (int i = 0; i < 3; ++i) {
        const int d = threadIdx.x + i * 128;
        const float yn = (v[i] - mu) * rstd * lng[d] + lnb[d];
        yg[(b * LL + s) * DI + d] = yn * zs[(b * LL + s) * DI + d];
    }
}

// ---------------------------------------------------------------------------
// Kernel 6: out_proj GEMM  (4096 x 384) @ (384 x 192) -> d_out
// ---------------------------------------------------------------------------
__global__ __launch_bounds__(256) void k_outproj(const float* __restrict__ yg,
                                                 const float* __restrict__ Wo,
                                                 float* __restrict__ out) {
    const int lane = threadIdx.x & 31;
    const int gw = blockIdx.x * 8 + (threadIdx.x >> 5);
    const int tm = gw / 12;                 // 192/16 = 12 N-tiles
    const int tn = gw % 12;
    const int m = tm * 16 + (lane & 15);
    const int n = tn * 16 + (lane & 15);
    const int khalf = (lane >> 4) * 2;
    const float* arow = yg + m * DI;
    const float* brow = Wo + n * DI;
    v8f acc = {};
    for (int kk = 0; kk < DI; kk += 4) {
        v2f a, bf;
        a.x = arow[kk + khalf];
        a.y = arow[kk + khalf + 1];
        bf.x = brow[kk + khalf];
        bf.y = brow[kk + khalf + 1];
        acc = __builtin_amdgcn_wmma_f32_16x16x4_f32(false, a, false, bf,
                                                    (short)0, acc, false, false);
    }
    const int mbase = tm * 16 + ((lane >> 4) << 3);
#pragma unroll
    for (int r = 0; r < 8; ++r)
        out[(mbase + r) * DMO + n] = acc[r];
}

// ---------------------------------------------------------------------------
extern "C" void kernel_launch(void* const* d_in, const int* in_sizes, int n_in,
                              void* d_out, int out_size, void* d_ws, size_t ws_size,
                              hipStream_t stream) {
    (void)in_sizes; (void)n_in; (void)out_size; (void)ws_size;
    const float* x    = (const float*)d_in[0];
    const float* ipw  = (const float*)d_in[1];
    const float* cw   = (const float*)d_in[2];
    const float* cb   = (const float*)d_in[3];
    const float* xpw  = (const float*)d_in[4];
    const float* dtw  = (const float*)d_in[5];
    const float* dtb  = (const float*)d_in[6];
    const float* alog = (const float*)d_in[7];
    const float* Dsp  = (const float*)d_in[8];
    const float* lng  = (const float*)d_in[9];
    const float* lnb  = (const float*)d_in[10];
    const float* opw  = (const float*)d_in[11];
    float* out = (float*)d_out;

    float* ws   = (float*)d_ws;
    float* xi   = ws;                        // (B, L, DI)
    float* zs   = xi + BB * LL * DI;         // (B, L, DI)  silu(z)
    float* xc   = zs + BB * LL * DI;         // (B, DI, L)
    float* xdbl = xc + BB * DI * LL;         // (B*K, 48, L)
    float* ys   = xdbl + BB * KD * CPAD * LL;// (B*K, DI, L)
    float* yg   = ys + BB * KD * DI * LL;    // (B, L, DI)

    k_inproj<<<1536, 256, 0, stream>>>(x, ipw, xi, zs);                 // 12288 waves
    k_conv<<<(BB * DI * LL + 255) / 256, 256, 0, stream>>>(xi, cw, cb, xc);
    k_xdbl<<<384, 256, 0, stream>>>(xpw, xc, xdbl);                     // 3072 waves
    k_scan<<<BB * KD * 3, 128, 0, stream>>>(xdbl, xc, dtw, dtb, alog, Dsp, ys);
    k_merge_ln<<<BB * LL, 128, 0, stream>>>(ys, zs, lng, lnb, yg);
    k_outproj<<<384, 256, 0, stream>>>(yg, opw, out);                   // 3072 waves
}